// DeepWelchTransform_26259430048288
// MI455X (gfx1250) — compile-verified
//
#include <hip/hip_runtime.h>
#include <math.h>

typedef __attribute__((ext_vector_type(2))) float v2f;
typedef __attribute__((ext_vector_type(8))) float v8f;

#define B_ROWS   1024
#define SEQ      16384
#define STEP_    256
#define NWIN_    61
#define NGRP     (SEQ / 4)        // 4096 K-groups of 4

// d_ws layout (floats):
//   [0, 2048)             : per-row accumulators, (re, im) interleaved
//   [2048, 2048 + 262144) : pre-swizzled B-fragment table:
//                           btab[g][lane] = (b.x, b.y) register pair for K-group g
//                           per ISA 7.12.2 B-layout (VGPR v: K=v lanes 0-15, K=v+2 lanes 16-31)

// Kernel 1: zero accumulators and build the pre-swizzled B-fragment table.
// Collapsed coefficient: C[s] = (1/61) * sum_{w: 0<=w<=60, p=s-256w in [0,1024)} {cos,-sin}(2*pi*f[p]).
__global__ void dwt_prep_kernel(const float* __restrict__ freqs,
                                float* __restrict__ acc,
                                float2* __restrict__ btab) {
  const int t = blockIdx.x * blockDim.x + threadIdx.x;   // 0 .. 131071
  if (t < 2 * B_ROWS) acc[t] = 0.0f;                     // ws is poisoned; re-zero each call

  const int g    = t >> 5;                               // K-group (k0 = 4g)
  const int L    = t & 31;                               // lane slot
  const int n    = L & 15;                               // B column
  const int half = L >> 4;

  float bx = 0.0f, by = 0.0f;
  if (n < 2) {                                           // only columns 0 (re) / 1 (im) nonzero
    const float twopi = 6.283185307179586f;
    const float scale = 1.0f / (float)NWIN_;
    const int kidx = 4 * g + 2 * half;                   // samples kidx (-> b.x) and kidx+1 (-> b.y)
#pragma unroll
    for (int e = 0; e < 2; ++e) {
      const int s  = kidx + e;
      const int r  = s & (STEP_ - 1);
      const int wb = s >> 8;
      float cacc = 0.0f, sacc = 0.0f;
#pragma unroll
      for (int j = 0; j < 4; ++j) {
        const int w = wb - j;
        if (w >= 0 && w <= NWIN_ - 1) {
          float sv, cv;
          sincosf(twopi * freqs[r + STEP_ * j], &sv, &cv);
          cacc += cv;
          sacc += sv;
        }
      }
      const float val = ((n == 0) ? cacc : -sacc) * scale;  // col 0: cos, col 1: -sin
      if (e == 0) bx = val; else by = val;
    }
  }
  btab[t] = make_float2(bx, by);
}

// Kernel 2: D(16x16) += A(16x4 f32) x B(4x16), streaming K.
// 4096 waves: rowgrp = W>>6 (16 rows), kchunk = W&63 (256 K values, 64 WMMA steps).
// Inner loop iteration: 4 unconditional global_load_b64 + 2 v_wmma (c0/c1), immediate-offset addressing.
__global__ void __launch_bounds__(256)
dwt_wmma_kernel(const float* __restrict__ input,
                const v2f* __restrict__ btab,
                float* __restrict__ acc) {
  const int lane = threadIdx.x & 31;
  const int wave = threadIdx.x >> 5;
  const int W = blockIdx.x * 8 + wave;     // 0..4095
  const int rowgrp = W >> 6;               // 0..63
  const int kchunk = W & 63;               // 0..63
  const int row0 = rowgrp << 4;
  const int kbase = kchunk << 8;
  const int n    = lane & 15;              // A row index == B/D column index
  const int half = lane >> 4;

  // A fragment base: lane holds input[row0+n, kidx..kidx+1], kidx = k0 + 2*half
  const v2f* ap = (const v2f*)(input + (size_t)(row0 + n) * SEQ) + (kbase >> 1) + half;
  // B fragment base: coalesced, one v2f per lane per K-group
  const v2f* bp = btab + (size_t)(kbase >> 2) * 32 + lane;

  v8f c0 = {};
  v8f c1 = {};
#pragma unroll 16
  for (int i = 0; i < 64; i += 2) {
    const v2f a0 = ap[i * 2];              // +16B per step
    const v2f b0 = bp[i * 32];             // +256B per step
    const v2f a1 = ap[i * 2 + 2];
    const v2f b1 = bp[i * 32 + 32];
    c0 = __builtin_amdgcn_wmma_f32_16x16x4_f32(false, a0, false, b0,
                                               (short)0, c0, false, false);
    c1 = __builtin_amdgcn_wmma_f32_16x16x4_f32(false, a1, false, b1,
                                               (short)0, c1, false, false);
  }

  // D layout: lane (n + 16*half), VGPR v holds D[v + 8*half, n].
  // Columns 0/1 carry re/im partials; reduce K-chunks via f32 atomics.
  if (n < 2) {
    const int rbase = row0 + 8 * half;
#pragma unroll
    for (int v = 0; v < 8; ++v) {
      atomicAdd(&acc[(rbase + v) * 2 + n], c0[v] + c1[v]);
    }
  }
}

// Kernel 3: psd -> linear layer.
__global__ void dwt_finalize_kernel(const float* __restrict__ acc,
                                    const float* __restrict__ fc_w,
                                    const float* __restrict__ fc_b,
                                    float* __restrict__ out) {
  const int b = blockIdx.x * blockDim.x + threadIdx.x;
  if (b < B_ROWS) {
    const float fr = acc[2 * b];
    const float fi = acc[2 * b + 1];
    out[b] = (fr * fr + fi * fi) * fc_w[0] + fc_b[0];
  }
}

extern "C" void kernel_launch(void* const* d_in, const int* in_sizes, int n_in,
                              void* d_out, int out_size, void* d_ws, size_t ws_size,
                              hipStream_t stream) {
  const float* input = (const float*)d_in[0];   // (1024, 16384) f32
  const float* freqs = (const float*)d_in[1];   // (1024,) f32
  const float* fc_w  = (const float*)d_in[2];   // (1,1) f32
  const float* fc_b  = (const float*)d_in[3];   // (1,) f32
  float* out = (float*)d_out;                   // (1024, 1) f32
  float* ws  = (float*)d_ws;

  float*  acc  = ws;
  float2* btab = (float2*)(ws + 2 * B_ROWS);    // 1 MB table

  dwt_prep_kernel<<<(NGRP * 32) / 256, 256, 0, stream>>>(freqs, acc, btab);
  dwt_wmma_kernel<<<512, 256, 0, stream>>>(input, (const v2f*)btab, acc);
  dwt_finalize_kernel<<<(B_ROWS + 255) / 256, 256, 0, stream>>>(acc, fc_w, fc_b, out);
}